// MessageAttention_82952998355585
// MI455X (gfx1250) — compile-verified
//
#include <hip/hip_runtime.h>

// Problem constants (fixed by the reference): H=128, A=8, D=16.
#define HDIM 128
#define AHEADS 8
#define DHEAD 16

typedef float v2f __attribute__((ext_vector_type(2)));
typedef float v8f __attribute__((ext_vector_type(8)));

// ---------------------------------------------------------------------------
// Zero a float buffer (denominators + output accumulators).
// ---------------------------------------------------------------------------
__global__ void zero_f32(float* __restrict__ p, int n) {
  int i = blockIdx.x * blockDim.x + threadIdx.x;
  if (i < n) p[i] = 0.0f;
}

// ---------------------------------------------------------------------------
// Per-node projection GEMM:  Y[N,128] = X[N,128] @ W^T[128,128] + b
// using V_WMMA_F32_16X16X4_F32 (exact f32, matches reference numerics).
//
// One wave computes a 16-row x 128-col tile: 8 accumulators (v8f each),
// K-loop of 32 steps of 4. Fragment layouts (wave32, ISA 7.12.2):
//   A 16x4  f32: lane l<16 holds A[l][k0+0..1], lane l+16 holds A[l][k0+2..3]
//   B 4x16  f32: lane l<16 holds B[k0+0..1][l], lane l+16 holds B[k0+2..3][l]
//   C/D 16x16:   VGPR r -> M=r (lanes 0-15), M=r+8 (lanes 16-31), N=lane&15
// With B[k][n] = W[h0+n][k], both A and B loads are one float2 per lane at
//   base + (lane&15)*128 + k0 + 2*(lane>>4).
// blockIdx.y selects which of the three projections (q/k/v) this block does.
// ---------------------------------------------------------------------------
__global__ __launch_bounds__(128) void proj_gemm(
    const float* __restrict__ q,  const float* __restrict__ k,  const float* __restrict__ v,
    const float* __restrict__ Wq, const float* __restrict__ Wk, const float* __restrict__ Wv,
    const float* __restrict__ bq, const float* __restrict__ bk, const float* __restrict__ bv,
    float* __restrict__ Qn, float* __restrict__ Kn, float* __restrict__ Vn, int N) {
  const float* X; const float* W; const float* b; float* Y;
  switch (blockIdx.y) {
    case 0:  X = q; W = Wq; b = bq; Y = Qn; break;
    case 1:  X = k; W = Wk; b = bk; Y = Kn; break;
    default: X = v; W = Wv; b = bv; Y = Vn; break;
  }

  const int lane = threadIdx.x & 31;
  const int wave = threadIdx.x >> 5;
  const int l15  = lane & 15;
  const int half = lane >> 4;            // 0: lanes 0-15, 1: lanes 16-31

  const int rowTile = blockIdx.x * 4 + wave;   // 4 waves per block
  const int row0 = rowTile * 16;

  // Clamp A-row for tail tiles: garbage rows are computed but never stored,
  // keeping EXEC all-ones for WMMA.
  int arow = row0 + l15;
  if (arow > N - 1) arow = N - 1;

  const float* __restrict__ xp = X + (size_t)arow * HDIM + 2 * half;
  const float* __restrict__ wp = W + (size_t)l15  * HDIM + 2 * half;

  v8f acc[8] = {};

  for (int k0 = 0; k0 < HDIM; k0 += 4) {
    v2f afrag = *(const v2f*)(xp + k0);
#pragma unroll
    for (int j = 0; j < 8; ++j) {
      v2f bfrag = *(const v2f*)(wp + j * (16 * HDIM) + k0);
      // 8 args: (neg_a, A, neg_b, B, c_mod, C, reuse_a, reuse_b)
      acc[j] = __builtin_amdgcn_wmma_f32_16x16x4_f32(
          false, afrag, false, bfrag, (short)0, acc[j], false, false);
    }
  }

#pragma unroll
  for (int j = 0; j < 8; ++j) {
    const float bias = b[j * 16 + l15];
#pragma unroll
    for (int r = 0; r < 8; ++r) {
      const int m = row0 + r + 8 * half;
      if (m < N) Y[(size_t)m * HDIM + j * 16 + l15] = acc[j][r] + bias;
    }
  }
}

// ---------------------------------------------------------------------------
// Pass 1: per-(edge, head) score -> exp -> atomic add into denom[N,8].
// Thread t handles edge e = t/8, head a = t%8; 8 consecutive threads cover
// one contiguous 512B projected row, so gathers stay coalesced within L2.
// ---------------------------------------------------------------------------
__global__ void edge_score(const float* __restrict__ Qn, const float* __restrict__ Kn,
                           const int* __restrict__ qidx, const int* __restrict__ kidx,
                           float* __restrict__ denom, int E) {
  int t = blockIdx.x * blockDim.x + threadIdx.x;
  if (t >= E * AHEADS) return;
  const int e = t >> 3;
  const int a = t & 7;
  const int dq = qidx[e];
  const int dk = kidx[e];

  const float4* __restrict__ qp = (const float4*)(Qn + (size_t)dq * HDIM + a * DHEAD);
  const float4* __restrict__ kp = (const float4*)(Kn + (size_t)dk * HDIM + a * DHEAD);

  float s = 0.0f;
#pragma unroll
  for (int i = 0; i < 4; ++i) {
    float4 qv = qp[i];
    float4 kv = kp[i];
    s += qv.x * kv.x + qv.y * kv.y + qv.z * kv.z + qv.w * kv.w;
  }
  const float ex = __expf(s * 0.25f);   // 1/sqrt(D), D=16

  unsafeAtomicAdd(&denom[(size_t)dq * AHEADS + a], ex);   // global_atomic_add_f32
}

// ---------------------------------------------------------------------------
// Pass 2: recompute exp(score) (L2 hits, cheaper than storing ex[E,8]),
// prob = ex / denom[dq,a], scatter prob * V into out[N,128] with fp32
// hardware atomics.
// ---------------------------------------------------------------------------
__global__ void edge_ctx(const float* __restrict__ Qn, const float* __restrict__ Kn,
                         const float* __restrict__ Vn,
                         const int* __restrict__ qidx, const int* __restrict__ kidx,
                         const float* __restrict__ denom,
                         float* __restrict__ out, int E) {
  int t = blockIdx.x * blockDim.x + threadIdx.x;
  if (t >= E * AHEADS) return;
  const int e = t >> 3;
  const int a = t & 7;
  const int dq = qidx[e];
  const int dk = kidx[e];

  const float4* __restrict__ qp = (const float4*)(Qn + (size_t)dq * HDIM + a * DHEAD);
  const float4* __restrict__ kp = (const float4*)(Kn + (size_t)dk * HDIM + a * DHEAD);

  float s = 0.0f;
#pragma unroll
  for (int i = 0; i < 4; ++i) {
    float4 qv = qp[i];
    float4 kv = kp[i];
    s += qv.x * kv.x + qv.y * kv.y + qv.z * kv.z + qv.w * kv.w;
  }
  const float prob = __expf(s * 0.25f) / denom[(size_t)dq * AHEADS + a];

  const float4* __restrict__ vp = (const float4*)(Vn + (size_t)dk * HDIM + a * DHEAD);
  float* __restrict__ op = out + (size_t)dq * HDIM + a * DHEAD;
#pragma unroll
  for (int i = 0; i < 4; ++i) {
    float4 vv = vp[i];
    unsafeAtomicAdd(op + 4 * i + 0, prob * vv.x);
    unsafeAtomicAdd(op + 4 * i + 1, prob * vv.y);
    unsafeAtomicAdd(op + 4 * i + 2, prob * vv.z);
    unsafeAtomicAdd(op + 4 * i + 3, prob * vv.w);
  }
}

// ---------------------------------------------------------------------------
// Host launcher.
// Inputs (setup_inputs order): q, k, v, Wq, bq, Wk, bk, Wv, bv, q_indices,
// k_indices. Workspace: Qn | Kn | Vn (N*128 f32 each) | denom (N*8 f32),
// total ~78.4 MB.
// ---------------------------------------------------------------------------
extern "C" void kernel_launch(void* const* d_in, const int* in_sizes, int n_in,
                              void* d_out, int out_size, void* d_ws, size_t ws_size,
                              hipStream_t stream) {
  const float* q  = (const float*)d_in[0];
  const float* k  = (const float*)d_in[1];
  const float* v  = (const float*)d_in[2];
  const float* Wq = (const float*)d_in[3];
  const float* bq = (const float*)d_in[4];
  const float* Wk = (const float*)d_in[5];
  const float* bk = (const float*)d_in[6];
  const float* Wv = (const float*)d_in[7];
  const float* bv = (const float*)d_in[8];
  const int* qidx = (const int*)d_in[9];
  const int* kidx = (const int*)d_in[10];

  const int N = in_sizes[0] / HDIM;
  const int E = in_sizes[10];

  float* Qn    = (float*)d_ws;
  float* Kn    = Qn + (size_t)N * HDIM;
  float* Vn    = Kn + (size_t)N * HDIM;
  float* denom = Vn + (size_t)N * HDIM;
  float* out   = (float*)d_out;

  // Zero accumulators (captured in the graph -> re-zeroed every replay).
  {
    int n = N * AHEADS;
    zero_f32<<<(n + 255) / 256, 256, 0, stream>>>(denom, n);
    n = N * HDIM;
    zero_f32<<<(n + 255) / 256, 256, 0, stream>>>(out, n);
  }

  // Projections: one 16-row tile per wave, 4 waves per block, grid.y = q/k/v.
  {
    const int rowTiles = (N + 15) / 16;
    dim3 grid((rowTiles + 3) / 4, 3);
    proj_gemm<<<grid, 128, 0, stream>>>(q, k, v, Wq, Wk, Wv, bq, bk, bv,
                                        Qn, Kn, Vn, N);
  }

  // Edge passes: one thread per (edge, head).
  {
    const int total = E * AHEADS;
    const int blocks = (total + 255) / 256;
    edge_score<<<blocks, 256, 0, stream>>>(Qn, Kn, qidx, kidx, denom, E);
    edge_ctx<<<blocks, 256, 0, stream>>>(Qn, Kn, Vn, qidx, kidx, denom, out, E);
  }
}